// CausalAttention_30064771072317
// MI455X (gfx1250) — compile-verified
//
#include <hip/hip_runtime.h>
#include <math.h>

// ---------------------------------------------------------------------------
// CDNA5 (gfx1250) fp32 WMMA causal attention.
//   L = 2048, H = 2048, 16 heads, d = 128, fp32 end to end.
// Matrix math: V_WMMA_F32_16X16X4_F32 (full fp32, matches reference).
// Data movement: GLOBAL_LOAD_ASYNC_TO_LDS_B128 double-buffered staging
// (ASYNCcnt + s_wait_asynccnt), so HBM/L2 traffic overlaps the WMMA stream.
// ---------------------------------------------------------------------------

typedef __attribute__((ext_vector_type(2))) float v2f;
typedef __attribute__((ext_vector_type(8))) float v8f;

static __device__ __forceinline__ v8f wmma_f32(v2f a, v2f b, v8f c) {
  // D = A(16x4) * B(4x16) + C(16x16), fp32
  return __builtin_amdgcn_wmma_f32_16x16x4_f32(
      /*neg_a=*/false, a, /*neg_b=*/false, b,
      /*c_mod=*/(short)0, c, /*reuse_a=*/false, /*reuse_b=*/false);
}

// LDS byte offset of a __shared__ object = low 32 bits of its flat address.
static __device__ __forceinline__ unsigned lds_off(const void* p) {
  return (unsigned)(unsigned long long)p;
}

// Async copy of 16 bytes global -> LDS (GVS mode: sgpr base + vgpr offset).
// Tracked by ASYNCcnt; completion enforced with s_wait_asynccnt.
static __device__ __forceinline__ void async_b128(unsigned ldsaddr,
                                                  unsigned gbyteoff,
                                                  const void* base) {
  asm volatile("global_load_async_to_lds_b128 %0, %1, %2"
               :: "v"(ldsaddr), "v"(gbyteoff),
                  "s"((unsigned long long)base)
               : "memory");
}

static __device__ __forceinline__ void wait_async0() {
  asm volatile("s_wait_asynccnt 0x0" ::: "memory");
}

// ---------------------------------------------------------------------------
// GEMM: Out[M,N] = A[M,K] * W[N,K]^T   (Out[i,j] = dot(A[i,:], W[j,:]))
// Block: 256 threads = 8 waves arranged 4(M) x 2(N); block tile 128M x 128N.
// Wave tile 32M x 64N -> per K-step: 2 A-frags + 4 B-frags feed 8 WMMAs.
// K staged by 32 through double-buffered LDS filled with async copies.
// ---------------------------------------------------------------------------
#define GTM 128
#define GTN 128
#define GTK 32
#define GLP (GTK + 4)   // padded row stride (36): 16B-aligned rows, and the
                        // lane-strided B reads hit 16 distinct LDS banks.

__global__ __launch_bounds__(256)
void gemm_xwt_kernel(const float* __restrict__ A, const float* __restrict__ W,
                     float* __restrict__ Out, int M, int N, int K)
{
  __shared__ __align__(16) float As[2][GTM][GLP];
  __shared__ __align__(16) float Ws[2][GTN][GLP];

  const int mbase = blockIdx.x * GTM;
  const int nbase = blockIdx.y * GTN;
  const int tid   = threadIdx.x;
  const int wave  = tid >> 5, lane = tid & 31;
  const int wm    = wave >> 1;         // 0..3 -> 32-row strip
  const int wn    = wave & 1;          // 0..1 -> 64-col strip
  const int lhalf = lane >> 4, l16 = lane & 15;

  // Stage one K-slab (128x32 of A and of W): 1024 float4 each, 4/thread each.
  auto stage = [&](int k0, int buf) {
#pragma unroll
    for (int i = 0; i < 4; ++i) {
      int v   = tid + i * 256;         // float4 index
      int row = v >> 3;                // 8 float4 per row
      int c4  = v & 7;
      async_b128(lds_off(&As[buf][row][c4 * 4]),
                 (unsigned)((((size_t)(mbase + row) * K + k0 + c4 * 4)) * 4), A);
      async_b128(lds_off(&Ws[buf][row][c4 * 4]),
                 (unsigned)((((size_t)(nbase + row) * K + k0 + c4 * 4)) * 4), W);
    }
  };

  v8f c[2][4] = {};                    // 2 M-tiles x 4 N-tiles of 16x16

  stage(0, 0);
  wait_async0();
  __syncthreads();

  for (int k0 = 0; k0 < K; k0 += GTK) {
    const int buf = (k0 / GTK) & 1;
    if (k0 + GTK < K) stage(k0 + GTK, buf ^ 1);   // prefetch next slab

#pragma unroll
    for (int kk = 0; kk < GTK; kk += 4) {
      const int ka = kk + 2 * lhalf;   // lanes 16-31 hold K+2 per WMMA layout
      v2f a0 = *(const v2f*)&As[buf][wm * 32 + l16][ka];
      v2f a1 = *(const v2f*)&As[buf][wm * 32 + 16 + l16][ka];
#pragma unroll
      for (int t = 0; t < 4; ++t) {
        v2f b = *(const v2f*)&Ws[buf][wn * 64 + t * 16 + l16][ka]; // B[k][n]=W[n][k]
        c[0][t] = wmma_f32(a0, b, c[0][t]);
        c[1][t] = wmma_f32(a1, b, c[1][t]);
      }
    }
    wait_async0();                     // next slab landed in LDS
    __syncthreads();                   // all waves done reading current slab
  }

#pragma unroll
  for (int r = 0; r < 2; ++r) {
#pragma unroll
    for (int t = 0; t < 4; ++t) {
#pragma unroll
      for (int v = 0; v < 8; ++v) {
        int row = mbase + wm * 32 + r * 16 + v + 8 * lhalf;
        int col = nbase + wn * 64 + t * 16 + l16;
        Out[(size_t)row * N + col] = c[r][t][v];
      }
    }
  }
}

// ---------------------------------------------------------------------------
// RoPE applied in place to Q and K buffers laid out [L, H].
// Thread handles one (f, f+64) pair inside one head. buf 0 = Q, buf 1 = K.
// ---------------------------------------------------------------------------
__global__ __launch_bounds__(256)
void rope_kernel(float* __restrict__ Q, float* __restrict__ Kb, int L, int H)
{
  const int d = 128, half = 64;
  int idx = blockIdx.x * blockDim.x + threadIdx.x;
  int pairsPerBuf = L * (H / 2);
  if (idx >= 2 * pairsPerBuf) return;

  float* buf = (idx < pairsPerBuf) ? Q : Kb;
  int r = idx % pairsPerBuf;
  int i = r / (H / 2);                 // sequence position
  int p = r % (H / 2);
  int h = p / half;                    // head
  int f = p % half;                    // 0..63

  float invf  = __powf(10000.0f, -(float)f / 64.0f);  // 10000^(-f/64)
  float theta = (float)i * invf;
  float cth   = cosf(theta), sth = sinf(theta);

  size_t j1 = (size_t)i * H + h * d + f;
  size_t j2 = j1 + half;
  float x1 = buf[j1], x2 = buf[j2];
  buf[j1] = x1 * cth - x2 * sth;
  buf[j2] = x2 * cth + x1 * sth;
}

// ---------------------------------------------------------------------------
// Flash attention, fp32 WMMA, async double-buffered K/V staging.
// Block = 128 threads (4 waves); block handles 64 query rows of one head.
// ---------------------------------------------------------------------------
#define AQ 64      // query rows per block
#define AKT 32     // keys per tile
#define DH 128     // head dim
#define ALP 132    // padded LDS row stride for Q/K/V tiles
#define PLP 36     // padded LDS row stride for P staging

__global__ __launch_bounds__(128)
void attn_kernel(const float* __restrict__ Qb, const float* __restrict__ Kb,
                 const float* __restrict__ Vb, float* __restrict__ Ob,
                 int L, int H)
{
  __shared__ __align__(16) float Qs[AQ][ALP];
  __shared__ __align__(16) float Ks[2][AKT][ALP];
  __shared__ __align__(16) float Vs[2][AKT][ALP];
  __shared__ __align__(16) float Ps[4][16][PLP];   // per-wave P staging

  const int h   = blockIdx.y;
  const int qb  = blockIdx.x * AQ;
  const int tid = threadIdx.x;
  const int wave = tid >> 5, lane = tid & 31;
  const int lhalf = lane >> 4, l16 = lane & 15;
  const float rs = 0.08838834764831845f;   // 1/sqrt(128)

  // Async-stage K and V tiles (32 x 128 each): 1024 float4 each, 8/thread.
  auto stageKV = [&](int kt, int buf) {
    const int kbase = kt * AKT;
#pragma unroll
    for (int i = 0; i < 8; ++i) {
      int v   = tid + i * 128;
      int row = v >> 5;                // 32 float4 per row
      int c4  = v & 31;
      unsigned goff =
          (unsigned)((((size_t)(kbase + row) * H + h * DH + c4 * 4)) * 4);
      async_b128(lds_off(&Ks[buf][row][c4 * 4]), goff, Kb);
      async_b128(lds_off(&Vs[buf][row][c4 * 4]), goff, Vb);
    }
  };

  // Async-stage the Q block tile (64 x 128): 2048 float4, 16/thread.
#pragma unroll
  for (int i = 0; i < 16; ++i) {
    int v   = tid + i * 128;
    int row = v >> 5;
    int c4  = v & 31;
    async_b128(lds_off(&Qs[row][c4 * 4]),
               (unsigned)((((size_t)(qb + row) * H + h * DH + c4 * 4)) * 4), Qb);
  }
  stageKV(0, 0);
  wait_async0();
  __syncthreads();

  v8f o[8] = {};                       // 16 rows x 128 feats = 8 C tiles
  float mrow[8], lrow[8];
#pragma unroll
  for (int v = 0; v < 8; ++v) { mrow[v] = -__builtin_inff(); lrow[v] = 0.f; }

  const int nkt = (qb + AQ) / AKT;     // key tiles covering keys 0..qb+63
  for (int kt = 0; kt < nkt; ++kt) {
    const int kbase = kt * AKT;
    const int buf = kt & 1;
    if (kt + 1 < nkt) stageKV(kt + 1, buf ^ 1);   // prefetch next K/V tile

    // ---- S = Q * K^T over full head dim (two 16x16 key tiles) ----
    v8f s0 = {}, s1 = {};
#pragma unroll
    for (int kk = 0; kk < DH; kk += 4) {
      const int ka = kk + 2 * lhalf;
      v2f a  = *(const v2f*)&Qs[wave * 16 + l16][ka];
      v2f b0 = *(const v2f*)&Ks[buf][l16][ka];        // B[k=f][n=key]=K[key][f]
      v2f b1 = *(const v2f*)&Ks[buf][16 + l16][ka];
      s0 = wmma_f32(a, b0, s0);
      s1 = wmma_f32(a, b1, s1);
    }

    // ---- causal mask + online softmax (per C-slot row) ----
#pragma unroll
    for (int v = 0; v < 8; ++v) {
      const int qr  = qb + wave * 16 + v + 8 * lhalf;   // this slot's q row
      const int k0c = kbase + l16;
      const int k1c = kbase + 16 + l16;
      float e0 = (k0c <= qr) ? s0[v] * rs : -__builtin_inff();
      float e1 = (k1c <= qr) ? s1[v] * rs : -__builtin_inff();
      float mx = fmaxf(e0, e1);
#pragma unroll
      for (int msk = 1; msk <= 8; msk <<= 1)
        mx = fmaxf(mx, __shfl_xor(mx, msk, 32));        // reduce 16-lane half
      float mnew = fmaxf(mrow[v], mx);
      float sc   = __expf(mrow[v] - mnew);              // 0 when mrow = -inf
      float p0 = __expf(e0 - mnew);
      float p1 = __expf(e1 - mnew);
      float sm = p0 + p1;
#pragma unroll
      for (int msk = 1; msk <= 8; msk <<= 1)
        sm += __shfl_xor(sm, msk, 32);
      lrow[v] = lrow[v] * sc + sm;
      mrow[v] = mnew;
#pragma unroll
      for (int t = 0; t < 8; ++t) o[t][v] *= sc;        // rescale O acc
      Ps[wave][v + 8 * lhalf][l16]      = p0;           // C-layout -> LDS
      Ps[wave][v + 8 * lhalf][16 + l16] = p1;
    }

    // ---- O += P * V  (K-dim = 32 keys, 8 feature tiles) ----
#pragma unroll
    for (int kk = 0; kk < AKT; kk += 4) {
      const int ka = kk + 2 * lhalf;
      v2f a = *(const v2f*)&Ps[wave][l16][ka];          // A[m=q][k=key]
#pragma unroll
      for (int t = 0; t < 8; ++t) {
        v2f b;                                          // B[k=key][n=f]
        b.x = Vs[buf][ka][t * 16 + l16];
        b.y = Vs[buf][ka + 1][t * 16 + l16];
        o[t] = wmma_f32(a, b, o[t]);
      }
    }

    wait_async0();                     // next K/V tile landed
    __syncthreads();                   // all waves done with current tile
  }

  // ---- normalize and store O in [L, H] layout ----
#pragma unroll
  for (int v = 0; v < 8; ++v) {
    float inv = 1.0f / lrow[v];
    int qr = qb + wave * 16 + v + 8 * lhalf;
#pragma unroll
    for (int t = 0; t < 8; ++t)
      Ob[(size_t)qr * H + h * DH + t * 16 + l16] = o[t][v] * inv;
  }
}

// ---------------------------------------------------------------------------
// Launch
// ---------------------------------------------------------------------------
extern "C" void kernel_launch(void* const* d_in, const int* in_sizes, int n_in,
                              void* d_out, int out_size, void* d_ws, size_t ws_size,
                              hipStream_t stream)
{
  const int L = 2048, H = 2048;
  const float* x  = (const float*)d_in[0];
  const float* Wq = (const float*)d_in[1];
  const float* Wk = (const float*)d_in[2];
  const float* Wv = (const float*)d_in[3];
  const float* Wo = (const float*)d_in[4];
  float* out = (float*)d_out;

  float* Qb = (float*)d_ws;                 // 4 x 16MB fp32 buffers in ws
  float* Kb = Qb + (size_t)L * H;
  float* Vb = Kb + (size_t)L * H;
  float* Ob = Vb + (size_t)L * H;

  dim3 gg(L / GTM, H / GTN);                // 16 x 16 blocks
  gemm_xwt_kernel<<<gg, 256, 0, stream>>>(x, Wq, Qb, L, H, H);
  gemm_xwt_kernel<<<gg, 256, 0, stream>>>(x, Wk, Kb, L, H, H);
  gemm_xwt_kernel<<<gg, 256, 0, stream>>>(x, Wv, Vb, L, H, H);

  int ropeThreads = L * H;                  // 2 bufs * L * H/2 pairs
  rope_kernel<<<(ropeThreads + 255) / 256, 256, 0, stream>>>(Qb, Kb, L, H);

  dim3 ga(L / AQ, 16);                      // 32 q-tiles x 16 heads
  attn_kernel<<<ga, 128, 0, stream>>>(Qb, Kb, Vb, Ob, L, H);

  gemm_xwt_kernel<<<gg, 256, 0, stream>>>(Ob, Wo, out, L, H, H);
}